// HEMALoss_20641612825292
// MI455X (gfx1250) — compile-verified
//
#include <hip/hip_runtime.h>
#include <hip/hip_bf16.h>
#include <math.h>

// Problem constants (from reference)
#define NN 512          // batch N
#define DD 256          // embed dim
#define PHYS_W 0.1f

typedef float v2f __attribute__((ext_vector_type(2)));
typedef float v8f __attribute__((ext_vector_type(8)));

// ---------------------------------------------------------------------------
// 1) Row L2-normalization: one wave (32 lanes) per row of a (NN x DD) matrix.
// ---------------------------------------------------------------------------
__global__ void hema_normalize_kernel(const float* __restrict__ z,
                                      float* __restrict__ out) {
    const int row  = blockIdx.x * (blockDim.x >> 5) + (threadIdx.x >> 5);
    const int lane = threadIdx.x & 31;
    const float* p = z + row * DD;
    float vals[DD / 32];
    float s = 0.0f;
#pragma unroll
    for (int t = 0; t < DD / 32; ++t) {
        vals[t] = p[lane + t * 32];
        s += vals[t] * vals[t];
    }
#pragma unroll
    for (int off = 16; off > 0; off >>= 1)
        s += __shfl_xor(s, off, 32);
    const float inv = 1.0f / sqrtf(s);
    float* q = out + row * DD;
#pragma unroll
    for (int t = 0; t < DD / 32; ++t)
        q[lane + t * 32] = vals[t] * inv;
}

// ---------------------------------------------------------------------------
// 2) WMMA GEMM: H = Z @ W (+ optional bias), Z:(NN x DD), W:(DD x DD) row-major.
//    Computes both ha = z_a@W1a + b1 and hb = z_b@W1b in one grid.
//    One wave computes one 16x16 output tile with V_WMMA_F32_16X16X4_F32.
//    Tiles: 32 (i) x 16 (n) per matrix = 512; total 1024 tiles.
// ---------------------------------------------------------------------------
__global__ void hema_mlp_gemm_kernel(const float* __restrict__ z_a,
                                     const float* __restrict__ z_b,
                                     const float* __restrict__ W1,
                                     const float* __restrict__ b1,
                                     float* __restrict__ ha,
                                     float* __restrict__ hb) {
    const int lane = threadIdx.x & 31;
    const int wave = threadIdx.x >> 5;
    const int tile = blockIdx.x * (blockDim.x >> 5) + wave;

    const bool second = tile >= 512;            // hb half
    const int  tt  = tile & 511;
    const int  i0  = (tt >> 4) << 4;            // 32 i-tiles
    const int  n0  = (tt & 15) << 4;            // 16 n-tiles

    const float* Z = second ? z_b : z_a;
    const float* W = second ? (W1 + DD * DD) : W1;   // W1b = rows [DD, 2*DD)
    float*       H = second ? hb : ha;

    const int r    = lane & 15;                 // row-in-tile for A / col for B
    const int koff = (lane >> 4) << 1;          // 0 or 2

    // A fragment source: row (i0+r), columns k+koff, k+koff+1
    const float* arow = Z + (i0 + r) * DD + koff;
    // B fragment source: B[kk][n] = W[(k+kk)*DD + n0 + n]
    const float* bptr = W + (size_t)koff * DD + n0 + r;

    v8f c = {};
    for (int k = 0; k < DD; k += 4) {
        v2f a = *(const v2f*)(arow + k);
        v2f b;
        b.x = bptr[k * DD];
        b.y = bptr[k * DD + DD];
        c = __builtin_amdgcn_wmma_f32_16x16x4_f32(
                false, a, false, b, (short)0, c, false, false);
    }

    const int m_base = (lane >> 4) << 3;        // 0 or 8
    const int n      = lane & 15;
    const float bias = second ? 0.0f : b1[n0 + n];   // fuse +b1 into ha only
#pragma unroll
    for (int m = 0; m < 8; ++m)
        H[(i0 + m_base + m) * DD + n0 + n] = c[m] + bias;
}

// ---------------------------------------------------------------------------
// 3) WMMA GEMM: sim = za_n @ zb_n^T, written into the logits output region.
//    C[i][j] = sum_k za_n[i][k] * zb_n[j][k].  1024 tiles (32x32).
// ---------------------------------------------------------------------------
__global__ void hema_sim_wmma_kernel(const float* __restrict__ zan,
                                     const float* __restrict__ zbn,
                                     float* __restrict__ sim_out) {
    const int lane = threadIdx.x & 31;
    const int wave = threadIdx.x >> 5;
    const int tile = blockIdx.x * (blockDim.x >> 5) + wave;
    const int i0 = (tile >> 5) << 4;
    const int j0 = (tile & 31) << 4;

    const int r    = lane & 15;
    const int koff = (lane >> 4) << 1;

    const float* arow = zan + (i0 + r) * DD + koff;
    const float* brow = zbn + (j0 + r) * DD + koff;   // B[kk][n] = zbn[j0+n][kk]

    v8f c = {};
    for (int k = 0; k < DD; k += 4) {
        v2f a = *(const v2f*)(arow + k);
        v2f b = *(const v2f*)(brow + k);
        c = __builtin_amdgcn_wmma_f32_16x16x4_f32(
                false, a, false, b, (short)0, c, false, false);
    }

    const int m_base = (lane >> 4) << 3;
    const int n      = lane & 15;
#pragma unroll
    for (int m = 0; m < 8; ++m)
        sim_out[(i0 + m_base + m) * NN + j0 + n] = c[m];
}

// ---------------------------------------------------------------------------
// 4) Fused correction + temperature scaling:
//    logits[i,j] = (sim[i,j] + 0.1*(sum_k relu(ha[i,k]+hb[j,k])*W2[k] + b2)) / T
//    Block = 16x16 (i,j) tile, 256 threads. LDS tiles padded to stride 257
//    (257 % 64 == 1) so the 16 distinct hb-rows hit 16 distinct banks.
//    Also emits the transposed logits (output #3 of the reference tuple).
// ---------------------------------------------------------------------------
__global__ void hema_correction_kernel(const float* __restrict__ ha,
                                       const float* __restrict__ hb,
                                       const float* __restrict__ W2,
                                       const float* __restrict__ b2,
                                       const float* __restrict__ logt,
                                       float* __restrict__ logits,
                                       float* __restrict__ logitsT) {
    __shared__ float sha[16][DD + 1];
    __shared__ float shb[16][DD + 1];
    __shared__ float sw2[DD];

    const int i0 = blockIdx.y << 4;
    const int j0 = blockIdx.x << 4;
    const int t  = threadIdx.x;          // 0..255

    // Coalesced tile loads: row c, col t of each 16xDD tile.
#pragma unroll 4
    for (int c = 0; c < 16; ++c) {
        sha[c][t] = ha[(i0 + c) * DD + t];
        shb[c][t] = hb[(j0 + c) * DD + t];
    }
    sw2[t] = W2[t];
    __syncthreads();

    const int ti = t >> 4;
    const int tj = t & 15;

    float acc = 0.0f;
#pragma unroll 8
    for (int k = 0; k < DD; ++k) {
        const float v = sha[ti][k] + shb[tj][k];
        acc = fmaf(fmaxf(v, 0.0f), sw2[k], acc);
    }

    const float T  = fminf(fmaxf(expf(logt[0]), 1e-4f), 100.0f);
    const int gi = i0 + ti, gj = j0 + tj;
    const float lg = (logits[gi * NN + gj] + PHYS_W * (acc + b2[0])) / T;
    logits [gi * NN + gj] = lg;
    logitsT[gj * NN + gi] = lg;
}

// ---------------------------------------------------------------------------
// 5) Row-wise logsumexp of a (NN x NN) matrix; one wave per row.
//    Column LSE is obtained by running this on logitsT.
// ---------------------------------------------------------------------------
__global__ void hema_lse_rows_kernel(const float* __restrict__ M,
                                     float* __restrict__ out) {
    const int row  = blockIdx.x * (blockDim.x >> 5) + (threadIdx.x >> 5);
    const int lane = threadIdx.x & 31;
    const float* p = M + row * NN;

    float mx = -INFINITY;
    for (int c = lane; c < NN; c += 32) mx = fmaxf(mx, p[c]);
#pragma unroll
    for (int off = 16; off > 0; off >>= 1)
        mx = fmaxf(mx, __shfl_xor(mx, off, 32));

    float s = 0.0f;
    for (int c = lane; c < NN; c += 32) s += expf(p[c] - mx);
#pragma unroll
    for (int off = 16; off > 0; off >>= 1)
        s += __shfl_xor(s, off, 32);

    if (lane == 0) out[row] = mx + logf(s);
}

// ---------------------------------------------------------------------------
// 6) Final reduction: loss = mean(lse_row - diag)/2 + mean(lse_col - diag)/2.
//    Also writes the clamped temperature (last output element).
// ---------------------------------------------------------------------------
__global__ void hema_loss_kernel(const float* __restrict__ logits,
                                 const float* __restrict__ lse_row,
                                 const float* __restrict__ lse_col,
                                 const float* __restrict__ logt,
                                 float* __restrict__ out_full) {
    __shared__ float red[NN];
    const int t = threadIdx.x;
    const float diag = logits[t * (NN + 1)];
    red[t] = (lse_row[t] - diag) + (lse_col[t] - diag);
    __syncthreads();
    for (int s = NN / 2; s > 0; s >>= 1) {
        if (t < s) red[t] += red[t + s];
        __syncthreads();
    }
    if (t == 0) {
        out_full[0] = red[0] / (2.0f * (float)NN);
        out_full[1 + 2 * NN * NN] =
            fminf(fmaxf(expf(logt[0]), 1e-4f), 100.0f);
    }
}

// ---------------------------------------------------------------------------
// Launcher.  d_out layout (floats): [loss(1) | logits(NN*NN) | logitsT(NN*NN)
//                                    | temperature(1)]
// Workspace (floats): zan | zbn | ha | hb | lse_row | lse_col  (~2.1 MB)
// ---------------------------------------------------------------------------
extern "C" void kernel_launch(void* const* d_in, const int* in_sizes, int n_in,
                              void* d_out, int out_size, void* d_ws, size_t ws_size,
                              hipStream_t stream) {
    const float* z_a  = (const float*)d_in[0];
    const float* z_b  = (const float*)d_in[1];
    const float* W1   = (const float*)d_in[2];
    const float* b1   = (const float*)d_in[3];
    const float* W2   = (const float*)d_in[4];
    const float* b2   = (const float*)d_in[5];
    const float* logt = (const float*)d_in[6];

    float* out     = (float*)d_out;
    float* logits  = out + 1;
    float* logitsT = out + 1 + NN * NN;

    float* ws      = (float*)d_ws;
    float* zan     = ws;
    float* zbn     = ws + NN * DD;
    float* ha      = ws + 2 * NN * DD;
    float* hb      = ws + 3 * NN * DD;
    float* lse_row = ws + 4 * NN * DD;
    float* lse_col = lse_row + NN;

    // 1) normalize rows (8 waves/block -> 8 rows/block -> 64 blocks)
    hema_normalize_kernel<<<NN / 8, 256, 0, stream>>>(z_a, zan);
    hema_normalize_kernel<<<NN / 8, 256, 0, stream>>>(z_b, zbn);

    // 2) ha/hb GEMMs: 1024 tiles, 4 waves (tiles) per 128-thread block
    hema_mlp_gemm_kernel<<<256, 128, 0, stream>>>(z_a, z_b, W1, b1, ha, hb);

    // 3) sim GEMM into logits region: 1024 tiles
    hema_sim_wmma_kernel<<<256, 128, 0, stream>>>(zan, zbn, logits);

    // 4) correction + /T, writes logits and logitsT
    dim3 cg(NN / 16, NN / 16);
    hema_correction_kernel<<<cg, 256, 0, stream>>>(ha, hb, W2, b2, logt,
                                                   logits, logitsT);

    // 5) logsumexp over rows (and over columns via the transpose copy)
    hema_lse_rows_kernel<<<NN / 8, 256, 0, stream>>>(logits,  lse_row);
    hema_lse_rows_kernel<<<NN / 8, 256, 0, stream>>>(logitsT, lse_col);

    // 6) scalar loss + temperature
    hema_loss_kernel<<<1, NN, 0, stream>>>(logits, lse_row, lse_col, logt, out);
}